// CSPlaqCouplingLayer_78855599555033
// MI455X (gfx1250) — compile-verified
//
#include <hip/hip_runtime.h>
#include <hip/hip_bf16.h>

typedef __attribute__((ext_vector_type(2))) float v2f;
typedef __attribute__((ext_vector_type(8))) float v8f;

#define NB 8
#define LDIM 64
#define TWO_PI_F 6.2831853071795864769f
#define INV_TWO_PI_F 0.15915494309189533577f

// Branchless tanh-approx GELU: 0.5*v*(1+tanh(u)) == v * sigmoid(2u)
//   sigmoid(2u) = 1 - 1/(exp(2u)+1); __expf is a single v_exp_f32 and
//   saturates correctly (exp->inf => 1, exp->0 => 0). No EXEC-mask branches.
__device__ __forceinline__ float gelu_tanh(float v) {
    const float c = 0.7978845608028654f;   // sqrt(2/pi)
    float u = c * (v + 0.044715f * v * v * v);
    float e = __expf(2.0f * u);
    return v * (1.0f - 1.0f / (e + 1.0f));
}

// ---------------------------------------------------------------------------
// Kernel 0: pack weights directly into WMMA B-fragment order.
// A chunk = 4 consecutive K values. For chunk q, lane l (h=l>>4, lm=l&15),
// N-tile nt: the lane needs the float2 pair  (W[k][n], W[k+1][n])  with
// k = 4q + 2h, n = nt*16+lm.  We store that pair at
//     Wf[q*64 + nt*32 + l]          (one global_load_b64 per fragment)
// conv1: K padded 18->20 (5 chunks), k = tap*2 + c   (zero-padded tail)
// conv2: K = 288 (72 chunks),        k = tap*32 + c  (n >= 25 zero-padded)
// Also zeroes per-sample logJ accumulators (done every call: deterministic).
// ---------------------------------------------------------------------------
__global__ __launch_bounds__(256) void pack_zero_kernel(
    const float* __restrict__ w1, const float* __restrict__ w2,
    v2f* __restrict__ W1f, v2f* __restrict__ W2f,
    float* __restrict__ logJ, int nSamples) {
    int i = blockIdx.x * blockDim.x + threadIdx.x;
    int stride = gridDim.x * blockDim.x;
    for (int t = i; t < 5 * 64; t += stride) {
        int q = t >> 6, slot = t & 63;
        int nt = slot >> 5, lane = slot & 31;
        int h = lane >> 4, lm = lane & 15;
        int n = nt * 16 + lm;
        int k = q * 4 + 2 * h;
        v2f p;
        p.x = (k + 0 < 18) ? w1[(n * 2 + ((k + 0) & 1)) * 9 + ((k + 0) >> 1)] : 0.0f;
        p.y = (k + 1 < 18) ? w1[(n * 2 + ((k + 1) & 1)) * 9 + ((k + 1) >> 1)] : 0.0f;
        W1f[t] = p;
    }
    for (int t = i; t < 72 * 64; t += stride) {
        int q = t >> 6, slot = t & 63;
        int nt = slot >> 5, lane = slot & 31;
        int h = lane >> 4, lm = lane & 15;
        int n = nt * 16 + lm;
        int k = q * 4 + 2 * h;
        v2f p;
        if (n < 25) {
            p.x = w2[(n * 32 + ((k + 0) & 31)) * 9 + ((k + 0) >> 5)];
            p.y = w2[(n * 32 + ((k + 1) & 31)) * 9 + ((k + 1) >> 5)];
        } else {
            p.x = 0.0f; p.y = 0.0f;
        }
        W2f[t] = p;
    }
    for (int t = i; t < nSamples; t += stride) logJ[t] = 0.0f;
}

// ---------------------------------------------------------------------------
// Fused kernel: one block = one sample * one 2-row strip.
//   Phase 0: frozen-masked cos/sin of 6 wrapped input rows -> LDS
//   Phase 1: conv1 (implicit GEMM, V_WMMA_F32_16X16X4_F32, K=20) + GELU -> LDS
//   Phase 2: conv2 (tap-major implicit GEMM, K=288) + bias -> LDS net_out
//   Phase 3: per-pixel rational-quadratic spline, fx + logJ reduction
// ---------------------------------------------------------------------------
__global__ __launch_bounds__(256) void fused_u1_coupling_kernel(
    const float* __restrict__ x,
    const float* __restrict__ b1, const float* __restrict__ b2,
    const v2f* __restrict__ W1f, const v2f* __restrict__ W2f,
    float* __restrict__ fx_out, float* __restrict__ logJ_out) {

    __shared__ float in_lds[6 * LDIM * 2];      // [row][col][ch]  (3 KB)
    __shared__ float h1_lds[4 * LDIM * 32];     // [pix][ch]       (32 KB)
    __shared__ float nout_lds[2 * LDIM * 32];   // [pix][ch]       (16 KB)
    __shared__ float red_lds[256];

    const int tid  = threadIdx.x;
    const int lane = tid & 31;
    const int wave = tid >> 5;
    const int half = lane >> 4;     // 0: lanes 0-15, 1: lanes 16-31
    const int lm   = lane & 15;

    const int s  = blockIdx.x >> 5;       // sample
    const int rb = blockIdx.x & 31;       // row-pair within sample
    const int r0 = rb * 2;
    const size_t sBase = (size_t)s * (LDIM * LDIM);

    // ---------------- Phase 0: staged masked cos/sin input -----------------
    for (int idx = tid; idx < 6 * LDIM; idx += 256) {
        int trow = idx >> 6, col = idx & 63;
        int gr = (r0 - 2 + trow + LDIM) & (LDIM - 1);
        float xv = x[sBase + gr * LDIM + col];
        int pat = (gr + 2 * col) % 3;                  // 2 == frozen
        float fz = (pat == 2) ? 1.0f : 0.0f;
        float sn, cs;
        __sincosf(xv, &sn, &cs);
        in_lds[idx * 2 + 0] = fz * cs;
        in_lds[idx * 2 + 1] = fz * sn;
    }
    __syncthreads();

    // ---------------- Phase 1: conv1 (2->32ch) via f32 WMMA ----------------
    // h1 block: 4 rows (x rows r0-1..r0+2) x 64 cols = 256 pixels, 8 waves x 32
    {
        const int mbase = wave * 32;
        v8f acc[2][2] = {};
        #pragma unroll
        for (int q = 0; q < 5; ++q) {
            // lane's tap = 2q + half; tail (k>=18) has zero B -> clamp tap,
            // branchless (real finite A value * zero weight).
            int tap = 2 * q + half;
            int tapc = (tap > 8) ? 8 : tap;
            int dy = tapc / 3, dxx = tapc - 3 * dy;
            v2f a[2];
            #pragma unroll
            for (int mt = 0; mt < 2; ++mt) {
                int pix = mbase + mt * 16 + lm;
                int hrow = pix >> 6, col = pix & 63;
                int base = ((hrow + dy) * LDIM + ((col + dxx + 63) & 63)) * 2;
                a[mt] = *(const v2f*)&in_lds[base];     // (cos, sin) pair
            }
            v2f b0 = W1f[q * 64 + lane];
            v2f b1 = W1f[q * 64 + 32 + lane];
            acc[0][0] = __builtin_amdgcn_wmma_f32_16x16x4_f32(
                false, a[0], false, b0, (short)0, acc[0][0], false, false);
            acc[0][1] = __builtin_amdgcn_wmma_f32_16x16x4_f32(
                false, a[0], false, b1, (short)0, acc[0][1], false, false);
            acc[1][0] = __builtin_amdgcn_wmma_f32_16x16x4_f32(
                false, a[1], false, b0, (short)0, acc[1][0], false, false);
            acc[1][1] = __builtin_amdgcn_wmma_f32_16x16x4_f32(
                false, a[1], false, b1, (short)0, acc[1][1], false, false);
        }
        #pragma unroll
        for (int nt = 0; nt < 2; ++nt) {
            int ch = nt * 16 + lm;
            float bias = b1[ch];
            #pragma unroll
            for (int mt = 0; mt < 2; ++mt)
                #pragma unroll
                for (int r = 0; r < 8; ++r) {
                    int pix = mbase + mt * 16 + r + 8 * half;
                    h1_lds[pix * 32 + ch] = gelu_tanh(acc[mt][nt][r] + bias);
                }
        }
    }
    __syncthreads();

    // ---------------- Phase 2: conv2 (32->25ch) via f32 WMMA ---------------
    // output: 2 rows x 64 = 128 pixels, 8 waves x 16 pixels, K = 288.
    // Tap-major: one LDS base per tap, 8 channel-chunks via immediate offsets.
    {
        const int mbase2 = wave * 16;
        const int pix = mbase2 + lm;
        const int orow = pix >> 6, col = pix & 63;
        v8f acc2[2] = {};
        #pragma unroll
        for (int tap = 0; tap < 9; ++tap) {
            const int dy = tap / 3, dxx = tap - 3 * dy;   // constants (unrolled)
            int abase = ((orow + dy) * LDIM + ((col + dxx + 63) & 63)) * 32
                        + 2 * half;
            const v2f* Bt = &W2f[tap * 8 * 64 + lane];
            #pragma unroll
            for (int cc = 0; cc < 8; ++cc) {
                v2f a  = *(const v2f*)&h1_lds[abase + 4 * cc];
                v2f b0 = Bt[cc * 64];
                v2f b1 = Bt[cc * 64 + 32];
                acc2[0] = __builtin_amdgcn_wmma_f32_16x16x4_f32(
                    false, a, false, b0, (short)0, acc2[0], false, false);
                acc2[1] = __builtin_amdgcn_wmma_f32_16x16x4_f32(
                    false, a, false, b1, (short)0, acc2[1], false, false);
            }
        }
        #pragma unroll
        for (int nt = 0; nt < 2; ++nt) {
            int ch = nt * 16 + lm;
            float bias = (ch < 25) ? b2[ch] : 0.0f;
            #pragma unroll
            for (int r = 0; r < 8; ++r) {
                int p = mbase2 + r + 8 * half;
                nout_lds[p * 32 + ch] = acc2[nt][r] + bias;
            }
        }
    }
    __syncthreads();

    // ---------------- Phase 3: rational-quadratic spline -------------------
    float lj = 0.0f;
    if (tid < 128) {
        int opix = tid, orow = opix >> 6, col = opix & 63;
        int gr = r0 + orow;
        float xv = x[sBase + gr * LDIM + col];
        int pat = (gr + 2 * col) % 3;    // 0=active, 1=passive, 2=frozen
        float fxv;
        if (pat == 0) {
            const float* o = &nout_lds[opix * 32];
            float t_shift = o[24];
            // softmax over widths / heights, softplus over derivatives
            float nw[NB], nh[NB], nd[NB];
            float mw = o[0], mh = o[NB];
            #pragma unroll
            for (int j = 1; j < NB; ++j) { mw = fmaxf(mw, o[j]); mh = fmaxf(mh, o[NB + j]); }
            float sw = 0.0f, sh = 0.0f;
            #pragma unroll
            for (int j = 0; j < NB; ++j) {
                nw[j] = __expf(o[j] - mw);          sw += nw[j];
                nh[j] = __expf(o[NB + j] - mh);     sh += nh[j];
                // branchless softplus: max(v,0) + log(1 + exp(-|v|))
                float dv = o[2 * NB + j];
                nd[j] = fmaxf(dv, 0.0f) + __logf(1.0f + __expf(-fabsf(dv)));
            }
            float iw = TWO_PI_F / sw, ih = TWO_PI_F / sh;
            // circular knots
            float kx[NB + 1], ky[NB + 1], sv[NB + 1];
            kx[0] = 0.0f; ky[0] = 0.0f;
            float cw = 0.0f, chh = 0.0f;
            #pragma unroll
            for (int j = 0; j < NB; ++j) {
                cw += nw[j] * iw;  kx[j + 1] = cw;
                chh += nh[j] * ih; ky[j + 1] = chh;
                sv[j] = nd[j];
            }
            sv[NB] = nd[0];
            float x1 = xv - floorf(xv * INV_TWO_PI_F) * TWO_PI_F;
            // register-resident bin select (unrolled; no scratch)
            float kxk = kx[0], kxk1 = kx[1], kyk = ky[0], kyk1 = ky[1];
            float sk = sv[0], sk1 = sv[1];
            #pragma unroll
            for (int j = 1; j < NB; ++j) {
                if (x1 >= kx[j]) {
                    kxk = kx[j]; kxk1 = kx[j + 1];
                    kyk = ky[j]; kyk1 = ky[j + 1];
                    sk = sv[j]; sk1 = sv[j + 1];
                }
            }
            float wk = kxk1 - kxk, hk = kyk1 - kyk;
            float slope = hk / wk;
            float xi = (x1 - kxk) / wk;
            xi = fminf(fmaxf(xi, 0.0f), 1.0f);
            float om = 1.0f - xi;
            float den = slope + (sk1 + sk - 2.0f * slope) * xi * om;
            float y = kyk + hk * (slope * xi * xi + sk * xi * om) / den;
            lj = __logf(slope * slope * (sk1 * xi * xi + 2.0f * slope * xi * om + sk * om * om))
                 - 2.0f * __logf(den);
            float f1 = y + t_shift;
            fxv = f1 - floorf(f1 * INV_TWO_PI_F) * TWO_PI_F;
        } else {
            fxv = xv;   // passive / frozen pass through
        }
        fx_out[sBase + gr * LDIM + col] = fxv;
    }
    // logJ block reduction + per-sample accumulate
    red_lds[tid] = lj;
    __syncthreads();
    for (int off = 128; off > 0; off >>= 1) {
        if (tid < off) red_lds[tid] += red_lds[tid + off];
        __syncthreads();
    }
    if (tid == 0) atomicAdd(&logJ_out[s], red_lds[0]);
}

// ---------------------------------------------------------------------------
// Launch
// ---------------------------------------------------------------------------
extern "C" void kernel_launch(void* const* d_in, const int* in_sizes, int n_in,
                              void* d_out, int out_size, void* d_ws, size_t ws_size,
                              hipStream_t stream) {
    const float* x  = (const float*)d_in[0];
    // d_in[1..3] = masks (recomputed analytically in-kernel)
    const float* w1 = (const float*)d_in[4];
    const float* b1 = (const float*)d_in[5];
    const float* w2 = (const float*)d_in[6];
    const float* b2 = (const float*)d_in[7];

    int nTotal = in_sizes[0];                 // N * 64 * 64
    int nSamples = nTotal / (LDIM * LDIM);    // 512

    float* out = (float*)d_out;
    float* fx  = out;
    float* lJ  = out + (size_t)nSamples * LDIM * LDIM;

    v2f* ws  = (v2f*)d_ws;
    v2f* W1f = ws;              // 5*64  float2  (2.5 KB)
    v2f* W2f = ws + 512;        // 72*64 float2  (36 KB)

    hipLaunchKernelGGL(pack_zero_kernel, dim3(40), dim3(256), 0, stream,
                       w1, w2, W1f, W2f, lJ, nSamples);
    hipLaunchKernelGGL(fused_u1_coupling_kernel, dim3(nSamples * 32), dim3(256), 0, stream,
                       x, b1, b2, W1f, W2f, fx, lJ);
}